// LineFinderLoss_78872779423805
// MI455X (gfx1250) — compile-verified
//
#include <hip/hip_runtime.h>
#include <math.h>

typedef __attribute__((ext_vector_type(2))) float v2f;
typedef __attribute__((ext_vector_type(8))) float v8f;

#define ALPHA 0.01f
#define EPSV  1e-5f
#define NB    16
#define NPRED 2048
#define NTARG 256

// ---------------------------------------------------------------------------
// Kernel 1: cost matrix via V_WMMA_F32_16X16X4_F32.
//   A (16x4)  = targ tile   (M = target rows)
//   B (4x16)  = inp^T tile  (N = pred cols)
//   D (16x16) = targ . inp^T dot products
// costT[t*2048 + p] = ALPHA*(|inp_p|^2 + |targ_t|^2 - 2*dot) - log(c_p+e) + log(1-c_p+e)
// One wave per 16x16 tile; 16 (targ) x 128 (pred) = 2048 tiles.
// ---------------------------------------------------------------------------
__global__ void cost_wmma_kernel(const float* __restrict__ pred,
                                 const float* __restrict__ label,
                                 float* __restrict__ costT) {
    const int wave = (blockIdx.x * blockDim.x + threadIdx.x) >> 5;
    const int lane = threadIdx.x & 31;
    const int t_tile = wave >> 7;        // 0..15
    const int p_tile = wave & 127;       // 0..127
    const int m0 = t_tile * 16;          // target row base
    const int n0 = p_tile * 16;          // pred  col base
    const int half = lane >> 4;          // 0 or 1 (K half)
    const int r    = lane & 15;

    const size_t predBase  = (size_t)(NB - 1) * NPRED * 5;
    const size_t labelBase = (size_t)(NB - 1) * NTARG * 4;

    // A: lane holds targ[m0+r][2*half], [2*half+1]
    const size_t la = labelBase + (size_t)(m0 + r) * 4 + 2 * half;
    v2f a; a.x = label[la]; a.y = label[la + 1];
    // B: lane holds inp[n0+r][2*half], [2*half+1]
    const size_t lb = predBase + (size_t)(n0 + r) * 5 + 2 * half;
    v2f b; b.x = pred[lb]; b.y = pred[lb + 1];

    v8f c = {};
    v8f d = __builtin_amdgcn_wmma_f32_16x16x4_f32(
        /*neg_a=*/false, a, /*neg_b=*/false, b,
        /*c_mod=*/(short)0, c, /*reuse_a=*/false, /*reuse_b=*/false);

    // Per-lane pred terms (N = r is fixed for this lane across all 8 D rows).
    const int p = n0 + r;
    const size_t pb = predBase + (size_t)p * 5;
    float i0 = pred[pb + 0], i1 = pred[pb + 1], i2 = pred[pb + 2], i3 = pred[pb + 3];
    float conf = pred[pb + 4];
    float nip = i0 * i0 + i1 * i1 + i2 * i2 + i3 * i3;
    float rowterm = -logf(conf + EPSV) + logf(1.0f - conf + EPSV);

#pragma unroll
    for (int k = 0; k < 8; ++k) {
        const int t = m0 + k + 8 * half;
        const size_t tb = labelBase + (size_t)t * 4;
        float g0 = label[tb + 0], g1 = label[tb + 1], g2 = label[tb + 2], g3 = label[tb + 3];
        float ntg = g0 * g0 + g1 * g1 + g2 * g2 + g3 * g3;
        float nd  = nip + ntg - 2.0f * d[k];
        costT[(size_t)t * NPRED + p] = ALPHA * nd + rowterm;
    }
}

// ---------------------------------------------------------------------------
// Kernel 2: shortest-augmenting-path LSA on cost (NTARG x NPRED), single
// cooperative workgroup (1024 threads = 32 waves), all state in LDS.
// Relax + argmin fused into one sweep: column j is owned by thread (j % 1024),
// so the post-relax shortest[j] is already in that thread's hands — no barrier
// needed between relax and argmin-candidate collection.
// ---------------------------------------------------------------------------
__global__ __launch_bounds__(1024)
void hungarian_kernel(const float* __restrict__ cost,
                      int* __restrict__ g_col4row,
                      int* __restrict__ g_row4col) {
    __shared__ float sh_short[NPRED];
    __shared__ float sh_v[NPRED];
    __shared__ int   sh_path[NPRED];
    __shared__ int   sh_row4col[NPRED];
    __shared__ unsigned char sh_SC[NPRED];
    __shared__ float sh_u[NTARG];
    __shared__ int   sh_col4row[NTARG];
    __shared__ unsigned char sh_SR[NTARG];
    __shared__ float red_val[32];
    __shared__ int   red_idx[32];
    __shared__ float s_minval;
    __shared__ int   s_i;
    __shared__ int   s_sink;

    const int tid  = threadIdx.x;
    const int nthr = blockDim.x;
    const int lane = tid & 31;
    const int wid  = tid >> 5;

    for (int j = tid; j < NPRED; j += nthr) { sh_v[j] = 0.0f; sh_row4col[j] = -1; }
    for (int r = tid; r < NTARG; r += nthr) { sh_u[r] = 0.0f; sh_col4row[r] = -1; }
    __syncthreads();

    for (int cur_row = 0; cur_row < NTARG; ++cur_row) {
        // per-row init
        for (int j = tid; j < NPRED; j += nthr) {
            sh_short[j] = INFINITY; sh_path[j] = -1; sh_SC[j] = 0;
        }
        for (int r = tid; r < NTARG; r += nthr) sh_SR[r] = 0;
        if (tid == 0) { s_minval = 0.0f; s_i = cur_row; s_sink = -1; }
        __syncthreads();

        // Dijkstra-style augmenting search
        for (;;) {
            const int   i    = s_i;
            const float minv = s_minval;
            if (tid == 0) sh_SR[i] = 1;
            const float base = minv - sh_u[i];
            const float* crow = cost + (size_t)i * NPRED;

            // fused relax + argmin candidate sweep over owned columns
            float bv = INFINITY; int bi = 0x7FFFFFFF;
            for (int j = tid; j < NPRED; j += nthr) {
                if (!sh_SC[j]) {
                    float dd = base + crow[j] - sh_v[j];
                    float sv = sh_short[j];
                    if (dd < sv) { sv = dd; sh_short[j] = dd; sh_path[j] = i; }
                    if (sv < bv || (sv == bv && j < bi)) { bv = sv; bi = j; }
                }
            }
            // wave-level argmin (tie-break: lowest index, matching np.argmin)
            for (int off = 16; off > 0; off >>= 1) {
                float ov = __shfl_down(bv, off);
                int   oi = __shfl_down(bi, off);
                if (ov < bv || (ov == bv && oi < bi)) { bv = ov; bi = oi; }
            }
            if (lane == 0) { red_val[wid] = bv; red_idx[wid] = bi; }
            __syncthreads();
            if (tid < 32) {
                float v2 = red_val[tid]; int i2 = red_idx[tid];
                for (int off = 16; off > 0; off >>= 1) {
                    float ov = __shfl_down(v2, off);
                    int   oi = __shfl_down(i2, off);
                    if (ov < v2 || (ov == v2 && oi < i2)) { v2 = ov; i2 = oi; }
                }
                if (tid == 0) {
                    s_minval = v2;
                    sh_SC[i2] = 1;
                    int rr = sh_row4col[i2];
                    if (rr < 0) s_sink = i2; else s_i = rr;
                }
            }
            __syncthreads();
            if (s_sink >= 0) break;
        }

        // dual updates (before augmentation, as in reference)
        const float mv = s_minval;
        if (tid == 0) sh_u[cur_row] += mv;
        for (int r = tid; r < NTARG; r += nthr) {
            if (sh_SR[r] && r != cur_row) sh_u[r] += mv - sh_short[sh_col4row[r]];
        }
        for (int j = tid; j < NPRED; j += nthr) {
            if (sh_SC[j]) sh_v[j] -= mv - sh_short[j];
        }
        __syncthreads();

        // augment along stored path (serial, thread 0)
        if (tid == 0) {
            int j = s_sink;
            for (;;) {
                int i = sh_path[j];
                sh_row4col[j] = i;
                int tmp = sh_col4row[i];
                sh_col4row[i] = j;
                j = tmp;
                if (i == cur_row) break;
            }
        }
        __syncthreads();
    }

    for (int r = tid; r < NTARG; r += nthr) g_col4row[r] = sh_col4row[r];
    for (int j = tid; j < NPRED; j += nthr) g_row4col[j] = sh_row4col[j];
}

// ---------------------------------------------------------------------------
// Kernel 3: final loss = ALPHA*sum matched |diff|^2 - sum matched log(c+e)
//                        - sum unmatched log(1-c+e)
// ---------------------------------------------------------------------------
__global__ __launch_bounds__(1024)
void loss_kernel(const float* __restrict__ pred,
                 const float* __restrict__ label,
                 const int* __restrict__ col4row,
                 const int* __restrict__ row4col,
                 float* __restrict__ out) {
    __shared__ float partial[32];
    const int tid  = threadIdx.x;
    const int lane = tid & 31;
    const int wid  = tid >> 5;
    const size_t predBase  = (size_t)(NB - 1) * NPRED * 5;
    const size_t labelBase = (size_t)(NB - 1) * NTARG * 4;

    float acc = 0.0f;

    // matched pairs (256)
    for (int t = tid; t < NTARG; t += blockDim.x) {
        int p = col4row[t];
        const size_t pb = predBase + (size_t)p * 5;
        const size_t tb = labelBase + (size_t)t * 4;
        float d0 = pred[pb + 0] - label[tb + 0];
        float d1 = pred[pb + 1] - label[tb + 1];
        float d2 = pred[pb + 2] - label[tb + 2];
        float d3 = pred[pb + 3] - label[tb + 3];
        float nd = d0 * d0 + d1 * d1 + d2 * d2 + d3 * d3;
        float conf = pred[pb + 4];
        acc += ALPHA * nd - logf(conf + EPSV);
    }
    // unmatched preds
    for (int p = tid; p < NPRED; p += blockDim.x) {
        if (row4col[p] < 0) {
            float conf = pred[predBase + (size_t)p * 5 + 4];
            acc += -logf(1.0f - conf + EPSV);
        }
    }

    for (int off = 16; off > 0; off >>= 1) acc += __shfl_down(acc, off);
    if (lane == 0) partial[wid] = acc;
    __syncthreads();
    if (tid < 32) {
        float v = partial[tid];
        for (int off = 16; off > 0; off >>= 1) v += __shfl_down(v, off);
        if (tid == 0) out[0] = v;
    }
}

// ---------------------------------------------------------------------------
extern "C" void kernel_launch(void* const* d_in, const int* in_sizes, int n_in,
                              void* d_out, int out_size, void* d_ws, size_t ws_size,
                              hipStream_t stream) {
    const float* pred  = (const float*)d_in[0];   // (16, 2048, 5)
    const float* label = (const float*)d_in[1];   // (16, 256, 4)
    float* out = (float*)d_out;                   // scalar f32

    float* costT   = (float*)d_ws;                                  // 256*2048 f32 (2 MB)
    int*   col4row = (int*)((char*)d_ws + (size_t)NTARG * NPRED * sizeof(float)); // 256
    int*   row4col = col4row + NTARG;                               // 2048

    // 2048 tiles, 8 waves (256 threads) per block -> 256 blocks
    cost_wmma_kernel<<<256, 256, 0, stream>>>(pred, label, costT);
    hungarian_kernel<<<1, 1024, 0, stream>>>(costT, col4row, row4col);
    loss_kernel<<<1, 1024, 0, stream>>>(pred, label, col4row, row4col, out);
}